// GRADEBase_26963804685186
// MI455X (gfx1250) — compile-verified
//
#include <hip/hip_runtime.h>

typedef float v2f __attribute__((ext_vector_type(2)));
typedef float v8f __attribute__((ext_vector_type(8)));

#define FEAT 128   // IN_DIM == HID_DIM == 128

// ---------------------------------------------------------------- degree prep
__global__ void k_fill1(float* __restrict__ p, int n) {
    int i = blockIdx.x * blockDim.x + threadIdx.x;
    if (i < n) p[i] = 1.0f;   // self-loop contributes 1 to the degree
}

__global__ void k_deg_count(const int* __restrict__ col, float* __restrict__ deg, int nE) {
    int e = blockIdx.x * blockDim.x + threadIdx.x;
    if (e < nE) unsafeAtomicAdd(&deg[col[e]], 1.0f);   // hw global_atomic_add_f32
}

__global__ void k_rsqrt(float* __restrict__ p, int n) {
    int i = blockIdx.x * blockDim.x + threadIdx.x;
    if (i < n) p[i] = rsqrtf(p[i]);
}

// ---------------------------------------------------------------- WMMA GEMM
// out[M x NDIM] = A[M x 128] @ W[128 x NDIM] (+ bias). A row stride lda (runtime).
// Block = 256 threads (8 waves) handles one 16-row m-tile, all n-tiles.
// W is staged in LDS with gfx1250 async global->LDS copies (ASYNCcnt), then each
// wave runs K=128 in steps of 4 via V_WMMA_F32_16X16X4_F32 with B fragments read
// from LDS at compile-time immediate offsets (NDIM templated).
template <int NDIM>
__global__ __launch_bounds__(256)
void k_wmma_gemm(const float* __restrict__ A, int lda, int M,
                 const float* __restrict__ W,
                 const float* __restrict__ bias,
                 float* __restrict__ out1, int ldo1,
                 float* __restrict__ out2, int ldo2)
{
    __shared__ float sW[128 * NDIM];                 // 64KB (NDIM=128) / 20KB (NDIM=40)

    // ---- async stage W into LDS: 128*NDIM floats, 16B per lane per issue.
    // 128*NDIM is a multiple of 1024 for NDIM in {128, 40}? 128*40=5120=5*1024 yes.
    {
        const int tid = threadIdx.x;
        const unsigned lbase = (unsigned)(size_t)&sW[0];
#pragma unroll 1
        for (int idx = tid * 4; idx < 128 * NDIM; idx += 256 * 4) {
            unsigned ldsAddr = lbase + (unsigned)idx * 4u;   // LDS byte address
            unsigned voff    = (unsigned)idx * 4u;           // global byte offset
            asm volatile("global_load_async_to_lds_b128 %0, %1, %2"
                         :: "v"(ldsAddr), "v"(voff), "s"(W) : "memory");
        }
        asm volatile("s_wait_asynccnt 0x0" ::: "memory");    // drain this wave's copies
    }
    __syncthreads();                                         // all waves' data visible

    const int lane = threadIdx.x & 31;
    const int wave = threadIdx.x >> 5;
    const int half = lane >> 4;          // 0: lanes 0-15, 1: lanes 16-31
    const int l16  = lane & 15;
    const int m0   = blockIdx.x * 16;
    const int NT   = (NDIM + 15) >> 4;

    // A fragment row for this lane (clamped for ragged M; stores are predicated)
    int arow = m0 + l16;
    if (arow >= M) arow = M - 1;
    const float* aptr = A + (size_t)arow * lda + half * 2;   // 8B aligned (lda even)

    for (int nt = wave; nt < NT; nt += 8) {                  // wave-uniform loop
        const int n0   = nt * 16;
        const int bcol = n0 + l16;
        const int bcr  = (bcol < NDIM) ? bcol : (NDIM - 1);  // clamp: cols >= NDIM never stored
        const float* wp = sW + half * 2 * NDIM + bcr;        // LDS, imm offsets below

        v8f c = {0.f, 0.f, 0.f, 0.f, 0.f, 0.f, 0.f, 0.f};
#pragma unroll
        for (int k = 0; k < FEAT; k += 4) {
            v2f a = *(const v2f*)(aptr + k);                 // A[m][ka], A[m][ka+1]
            v2f b;
            b.x = wp[k * NDIM];                              // ds_load_b32 offset:k*NDIM*4
            b.y = wp[(k + 1) * NDIM];
            c = __builtin_amdgcn_wmma_f32_16x16x4_f32(
                    /*neg_a=*/false, a, /*neg_b=*/false, b,
                    /*c_mod=*/(short)0, c, /*reuse_a=*/false, /*reuse_b=*/false);
        }

        if (bcol < NDIM) {
            const float bv = bias ? bias[bcol] : 0.f;
            const bool dup = (out2 != nullptr);
#pragma unroll
            for (int r = 0; r < 8; ++r) {
                const int rowi = m0 + r + half * 8;          // D: VGPR r -> M=r / r+8
                if (rowi < M) {
                    const float v = c[r] + bv;
                    out1[(size_t)rowi * ldo1 + bcol] = v;
                    if (dup) out2[(size_t)rowi * ldo2 + bcol] = v;
                }
            }
        }
    }
}

// ---------------------------------------------------------------- aggregation
// agg[i,:] = h[i,:] * dis[i]^2   (self-loop term; also initializes agg)
__global__ void k_selfloop(const float* __restrict__ h, const float* __restrict__ dis,
                           float* __restrict__ agg, int n) {
    int t = blockIdx.x * blockDim.x + threadIdx.x;
    int i = t >> 5;
    if (i >= n) return;
    int q = (t & 31) * 4;
    float d = dis[i];
    float s = d * d;
    float4 v = *(const float4*)(h + (size_t)i * FEAT + q);
    v.x *= s; v.y *= s; v.z *= s; v.w *= s;
    *(float4*)(agg + (size_t)i * FEAT + q) = v;
}

// agg[col[e],:] += h[row[e],:] * dis[row]*dis[col]   (one wave per edge, float4/lane)
__global__ void k_scatter(const int* __restrict__ row, const int* __restrict__ col,
                          const float* __restrict__ dis, const float* __restrict__ h,
                          float* __restrict__ agg, int nE) {
    int t = blockIdx.x * blockDim.x + threadIdx.x;
    int e = t >> 5;
    if (e >= nE) return;
    int q = (t & 31) * 4;
    int r = row[e], c = col[e];
    float norm = dis[r] * dis[c];
    float4 v = *(const float4*)(h + (size_t)r * FEAT + q);  // global_load_b128 (L2 resident)
    float* dst = agg + (size_t)c * FEAT + q;
    unsafeAtomicAdd(dst + 0, v.x * norm);
    unsafeAtomicAdd(dst + 1, v.y * norm);
    unsafeAtomicAdd(dst + 2, v.z * norm);
    unsafeAtomicAdd(dst + 3, v.w * norm);
}

// out[i, q..q+3] = relu(agg[i, q..q+3] + b[q..q+3]); out has row stride ld (=296)
__global__ void k_bias_relu(const float* __restrict__ agg, const float* __restrict__ b,
                            float* __restrict__ out, int ld, int n) {
    int t = blockIdx.x * blockDim.x + threadIdx.x;
    int i = t >> 5;
    if (i >= n) return;
    int q = (t & 31) * 4;
    float4 v = *(const float4*)(agg + (size_t)i * FEAT + q);
    float* o = out + (size_t)i * ld + q;    // stride 296 floats = 1184 B, a 16B multiple; scalar stores
    o[0] = fmaxf(v.x + b[q + 0], 0.f);
    o[1] = fmaxf(v.y + b[q + 1], 0.f);
    o[2] = fmaxf(v.z + b[q + 2], 0.f);
    o[3] = fmaxf(v.w + b[q + 3], 0.f);
}

// ---------------------------------------------------------------- launcher
extern "C" void kernel_launch(void* const* d_in, const int* in_sizes, int n_in,
                              void* d_out, int out_size, void* d_ws, size_t ws_size,
                              hipStream_t stream) {
    const float* x   = (const float*)d_in[0];
    const int*   ei  = (const int*)  d_in[1];   // [2, E] row-major
    const float* w0  = (const float*)d_in[2];
    const float* b0  = (const float*)d_in[3];
    const float* w1  = (const float*)d_in[4];
    const float* b1  = (const float*)d_in[5];
    const float* cw  = (const float*)d_in[6];
    const float* cb  = (const float*)d_in[7];

    const int n  = in_sizes[0] / FEAT;          // 100000
    const int nE = in_sizes[1] / 2;             // 1600000
    const int C  = 40;                          // N_CLASSES (reference)
    const int ldf = 2 * FEAT + C;               // 296

    const int* rowp = ei;
    const int* colp = ei + nE;

    float* ws  = (float*)d_ws;                  // needs ~103 MB
    float* dis = ws;                            // [n]
    float* h   = dis + n;                       // [n,128]
    float* agg = h + (size_t)n * FEAT;          // [n,128]

    float* logits = (float*)d_out;              // [n, C], stride C
    float* feat   = logits + (size_t)n * C;     // [n, 296]

    const int B = 256;
    const int gN  = (n + B - 1) / B;
    const int gE  = (nE + B - 1) / B;
    const int gN4 = (int)(((long)n * 32 + B - 1) / B);   // n*32 threads (float4 per lane)
    const int gE4 = (int)(((long)nE * 32 + B - 1) / B);  // nE*32 threads
    const int mtiles = (n + 15) / 16;

    // degrees (with self-loop) -> dis = rsqrt(deg)
    k_fill1    <<<gN, B, 0, stream>>>(dis, n);
    k_deg_count<<<gE, B, 0, stream>>>(colp, dis, nE);
    k_rsqrt    <<<gN, B, 0, stream>>>(dis, n);

    // ---- layer 1: h = x @ w0 ; agg = D^-1/2 (A+I) D^-1/2 h ; h1 = relu(agg+b0)
    k_wmma_gemm<128><<<mtiles, B, 0, stream>>>(x, FEAT, n, w0, nullptr, h, FEAT, nullptr, 0);
    k_selfloop <<<gN4, B, 0, stream>>>(h, dis, agg, n);
    k_scatter  <<<gE4, B, 0, stream>>>(rowp, colp, dis, h, agg, nE);
    k_bias_relu<<<gN4, B, 0, stream>>>(agg, b0, feat + 0, ldf, n);        // h1 -> feat[:,0:128]

    // ---- layer 2 (reads h1 from d_out feat region, stride 296)
    k_wmma_gemm<128><<<mtiles, B, 0, stream>>>(feat, ldf, n, w1, nullptr, h, FEAT, nullptr, 0);
    k_selfloop <<<gN4, B, 0, stream>>>(h, dis, agg, n);
    k_scatter  <<<gE4, B, 0, stream>>>(rowp, colp, dis, h, agg, nE);
    k_bias_relu<<<gN4, B, 0, stream>>>(agg, b1, feat + FEAT, ldf, n);     // h2 -> feat[:,128:256]

    // ---- classifier: logits = h2 @ cls_w + cls_b -> d_out[:,0:40] and feat[:,256:296]
    k_wmma_gemm<40><<<mtiles, B, 0, stream>>>(feat + FEAT, ldf, n, cw, cb,
                                              logits, C, feat + 2 * FEAT, ldf);
}